// ASCC_50216757625046
// MI455X (gfx1250) — compile-verified
//
#include <hip/hip_runtime.h>
#include <hip/hip_bf16.h>

typedef __attribute__((ext_vector_type(2))) float v2f;
typedef __attribute__((ext_vector_type(4))) float v4f;
typedef __attribute__((ext_vector_type(8))) float v8f;

#define IN_CH 64
#define OUT_CH 64
#define T_TAPS 16   // 4x4 after antisymmetrization
#define KDIM 1024   // T_TAPS * IN_CH

// ---------------------------------------------------------------------------
// Device-scope non-returning FP32 atomic add (guaranteed hardware atomic,
// no CAS fallback). Non-returning form -> STOREcnt path, RMW executes in L2.
// ---------------------------------------------------------------------------
__device__ __forceinline__ void gatomic_fadd_dev(float* p, float v) {
    asm volatile("global_atomic_add_f32 %0, %1, off scope:SCOPE_DEV"
                 :: "v"(p), "v"(v)
                 : "memory");
}

// ---------------------------------------------------------------------------
// Zero the accumulation buffer A [N*1024 floats], float4 grid-stride.
// ---------------------------------------------------------------------------
__global__ __launch_bounds__(256) void ascc_zero(v4f* __restrict__ p, long n4) {
    long i = (long)blockIdx.x * blockDim.x + threadIdx.x;
    long stride = (long)gridDim.x * blockDim.x;
    v4f z = {0.f, 0.f, 0.f, 0.f};
    for (; i < n4; i += stride) p[i] = z;
}

// ---------------------------------------------------------------------------
// Build antisymmetrized kernel Kf[1024,64] from kernel[4,2,64,64]:
//   kernel_full[x,y] = kernel[x,y]        for y < 2
//                    = -kernel[3-x,3-y]   for y >= 2
// tap t = x*4 + y ; flat row k = t*64 + i ; col o.
// ---------------------------------------------------------------------------
__global__ __launch_bounds__(256) void ascc_buildk(const float* __restrict__ kern,
                                                   float* __restrict__ Kf) {
    int idx = blockIdx.x * blockDim.x + threadIdx.x;   // 0 .. 65535
    int o = idx & 63;
    int k = idx >> 6;          // 0..1023
    int i = k & 63;
    int t = k >> 6;            // 0..15
    int x = t >> 2;
    int y = t & 3;
    float v;
    if (y < 2) {
        v = kern[(((x * 2 + y) * 64) + i) * 64 + o];
    } else {
        v = -kern[((((3 - x) * 2 + (3 - y)) * 64) + i) * 64 + o];
    }
    Kf[idx] = v;
}

// ---------------------------------------------------------------------------
// Edge scatter: 64 threads per edge (one per input channel), 4 edges/block.
// Coalesced 256B feature gather + 4 coalesced 256B atomic-add bursts.
// ---------------------------------------------------------------------------
__global__ __launch_bounds__(256) void ascc_scatter(const float* __restrict__ feats,
                                                    const int*   __restrict__ recv,
                                                    const float* __restrict__ relpos,
                                                    const float* __restrict__ wsup,
                                                    const int*   __restrict__ send,
                                                    float* __restrict__ A,
                                                    int E) {
    int tid = blockIdx.x * blockDim.x + threadIdx.x;
    int e = tid >> 6;
    int c = tid & 63;
    if (e >= E) return;

    float inv = 1.0f / wsup[0];
    float ux = relpos[2 * e + 0] * inv;
    float uy = relpos[2 * e + 1] * inv;
    ux = fminf(fmaxf(ux, -1.0f), 1.0f);
    uy = fminf(fmaxf(uy, -1.0f), 1.0f);

    float gx = (ux + 1.0f) * 0.5f * 3.0f;   // (kh-1) = 3
    float gy = (uy + 1.0f) * 0.5f * 3.0f;   // (kw-1) = 3
    float x0 = fminf(fmaxf(floorf(gx), 0.0f), 2.0f);
    float y0 = fminf(fmaxf(floorf(gy), 0.0f), 2.0f);
    float fx = gx - x0;
    float fy = gy - y0;
    int x0i = (int)x0;
    int y0i = (int)y0;

    float r2 = ux * ux + uy * uy;
    float tw = fmaxf(1.0f - r2, 0.0f);
    float win = tw * tw * tw;

    float f = feats[(long)send[e] * IN_CH + c] * win;

    float* base = A + ((long)recv[e] * T_TAPS + (x0i * 4 + y0i)) * IN_CH + c;
    gatomic_fadd_dev(base,       (1.0f - fx) * (1.0f - fy) * f);  // tap (x0  , y0  )
    gatomic_fadd_dev(base + 64,  (1.0f - fx) * fy          * f);  // tap (x0  , y0+1)
    gatomic_fadd_dev(base + 256, fx          * (1.0f - fy) * f);  // tap (x0+1, y0  )
    gatomic_fadd_dev(base + 320, fx          * fy          * f);  // tap (x0+1, y0+1)
}

// ---------------------------------------------------------------------------
// fp32 WMMA GEMM: out[M,64] = A[M,1024] * Kf[1024,64] + bias.
// One wave -> 16 rows x 64 cols (4 C-tiles). A fragment reused by 4 WMMAs.
// A 16x4 f32 layout: lanes 0-15 hold K={0,1}, lanes 16-31 hold K={2,3}
// B 4x16  f32 layout: lanes 0-15 hold rows K={kb,kb+1}, lanes 16-31 K={kb+2,kb+3}
// C 16x16 f32 layout: lane<16 -> M=vgpr, lane>=16 -> M=8+vgpr.
// A is streamed with NT temporal hint (read-once, > L2 capacity); Kf (256KB)
// stays cache-resident.
// ---------------------------------------------------------------------------
__global__ __launch_bounds__(256) void ascc_gemm(const float* __restrict__ A,
                                                 const float* __restrict__ Kf,
                                                 const float* __restrict__ bias,
                                                 float* __restrict__ out,
                                                 int M) {
    const int lane = threadIdx.x & 31;
    const int wave = blockIdx.x * (blockDim.x >> 5) + (threadIdx.x >> 5);
    const int row0 = wave * 16;
    if (row0 >= M) return;            // wave-uniform: EXEC stays all-ones

    const int half = lane >> 4;       // 0: K pair {0,1}; 1: K pair {2,3}
    const int l16  = lane & 15;

    int arow = row0 + l16;
    if (arow >= M) arow = M - 1;      // safe clamp (M=50000 is an exact multiple)

    const float* aptr = A + (long)arow * KDIM + half * 2;

    v8f acc[4] = {v8f{}, v8f{}, v8f{}, v8f{}};

    #pragma unroll 4
    for (int k0 = 0; k0 < KDIM; k0 += 4) {
        v2f afrag = __builtin_nontemporal_load((const v2f*)(aptr + k0));
        const int kb = k0 + half * 2;
        const float* bp = Kf + (long)kb * OUT_CH + l16;

        v2f b0 = {bp[0],  bp[0 + OUT_CH]};
        v2f b1 = {bp[16], bp[16 + OUT_CH]};
        v2f b2 = {bp[32], bp[32 + OUT_CH]};
        v2f b3 = {bp[48], bp[48 + OUT_CH]};

        acc[0] = __builtin_amdgcn_wmma_f32_16x16x4_f32(false, afrag, false, b0,
                                                       (short)0, acc[0], false, false);
        acc[1] = __builtin_amdgcn_wmma_f32_16x16x4_f32(false, afrag, false, b1,
                                                       (short)0, acc[1], false, false);
        acc[2] = __builtin_amdgcn_wmma_f32_16x16x4_f32(false, afrag, false, b2,
                                                       (short)0, acc[2], false, false);
        acc[3] = __builtin_amdgcn_wmma_f32_16x16x4_f32(false, afrag, false, b3,
                                                       (short)0, acc[3], false, false);
    }

    // Store D: lane<16 -> rows row0..row0+7, lane>=16 -> rows row0+8..row0+15.
    const int rbase = row0 + half * 8;
    #pragma unroll
    for (int j = 0; j < 4; ++j) {
        float bv = bias[j * 16 + l16];
        #pragma unroll
        for (int r = 0; r < 8; ++r) {
            int row = rbase + r;
            if (row < M) out[(long)row * OUT_CH + j * 16 + l16] = acc[j][r] + bv;
        }
    }
}

// ---------------------------------------------------------------------------
// Launch
// ---------------------------------------------------------------------------
extern "C" void kernel_launch(void* const* d_in, const int* in_sizes, int n_in,
                              void* d_out, int out_size, void* d_ws, size_t ws_size,
                              hipStream_t stream) {
    const float* features = (const float*)d_in[0];
    const int*   receivers = (const int*)d_in[1];
    const float* relpos   = (const float*)d_in[2];
    const float* wsup     = (const float*)d_in[3];
    const int*   senders  = (const int*)d_in[4];
    const float* kern     = (const float*)d_in[5];
    const float* bias     = (const float*)d_in[6];
    float* out = (float*)d_out;

    const int N = in_sizes[0] / IN_CH;   // 50000
    const int E = in_sizes[1];           // 800000

    float* A  = (float*)d_ws;                       // [N, 16, 64]
    float* Kf = A + (long)N * KDIM;                 // [1024, 64]

    // 1. zero accumulation buckets
    long n4 = (long)N * (KDIM / 4);
    ascc_zero<<<2048, 256, 0, stream>>>((v4f*)A, n4);

    // 2. antisymmetrized full kernel (16*64*64 = 65536 elems)
    ascc_buildk<<<(T_TAPS * IN_CH * OUT_CH) / 256, 256, 0, stream>>>(kern, Kf);

    // 3. edge scatter: 64 threads/edge, device-scope hardware FP atomics
    long sthreads = (long)E * IN_CH;
    int sblocks = (int)((sthreads + 255) / 256);
    ascc_scatter<<<sblocks, 256, 0, stream>>>(features, receivers, relpos, wsup,
                                              senders, A, E);

    // 4. fp32 WMMA contraction + bias
    int mtiles = (N + 15) / 16;                  // waves needed
    int gblocks = (mtiles + 7) / 8;              // 8 waves per 256-thread block
    ascc_gemm<<<gblocks, 256, 0, stream>>>(A, Kf, bias, out, N);
}